// PhysicsLoss_63548336112325
// MI455X (gfx1250) — compile-verified
//
#include <hip/hip_runtime.h>
#include <hip/hip_bf16.h>
#include <math.h>

typedef float v2f __attribute__((ext_vector_type(2)));
typedef float v8f __attribute__((ext_vector_type(8)));

// ---------------- workspace layout (floats) ----------------
// [0, 12800)      feat[128][100]   pooled branch features
// [12800, 12928)  b_u[128]
// [12928, 13056)  b_v[128]
// [13056, 13072)  acc[16]:
//   0..5  interior: SAu2, SAuBu, SBu2, SAv2, SAvBv, SBv2
//   6..7  dirichlet: Stu2, Stv2
//   8..13 forcing:  Scu2, Scucv, Scv2, Sdu2, Sdudv, Sdv2
//   14    forcing cross term  Sum_b bu*(R0+R2)+bv*(R1+R3)
//   15    Sum g^2
static const int WS_FEAT = 0;
static const int WS_BU   = 12800;
static const int WS_BV   = 12928;
static const int WS_ACC  = 13056;

#define NU_C   0.3f
#define PREF_C (200.0f / (1.0f - NU_C * NU_C))

// ---------------- block reduction (wave32) ----------------
__device__ __forceinline__ float blockReduceSum(float v, float* smem8) {
  __syncthreads();                       // protect smem reuse across calls
  #pragma unroll
  for (int o = 16; o > 0; o >>= 1) v += __shfl_xor(v, o, 32);
  const int lane = threadIdx.x & 31, wid = threadIdx.x >> 5;
  if (lane == 0) smem8[wid] = v;
  __syncthreads();
  float r = 0.f;
  if (wid == 0) {
    r = (lane < (int)(blockDim.x >> 5)) ? smem8[lane] : 0.f;
    #pragma unroll
    for (int o = 4; o > 0; o >>= 1) r += __shfl_xor(r, o, 32);
  }
  return r; // valid on thread 0
}

// ---------------- analytic trunk value / Jacobian / Hessian ----------------
// t = W3 tanh(W2 tanh(W1 x + b1) + b2) + b3, all dims = 2.
// J[m*2+k] = dt_m/dx_k ; H[m*3+{0,1,2}] = d2t_m/{dxdx, dxdy, dydy}
__device__ __forceinline__ void trunk_eval(
    const float* __restrict__ w1, const float* __restrict__ b1,
    const float* __restrict__ w2, const float* __restrict__ b2,
    const float* __restrict__ w3, const float* __restrict__ b3,
    float x0, float x1, float t[2], float J[4], float H[6]) {
  float h1[2], D1[2][2], H1[2][3];
  #pragma unroll
  for (int i = 0; i < 2; ++i) {
    float wa = w1[2 * i], wb = w1[2 * i + 1];
    float h = tanhf(wa * x0 + wb * x1 + b1[i]);
    float g = 1.f - h * h, cc = -2.f * h * g;
    h1[i] = h;
    D1[i][0] = g * wa;       D1[i][1] = g * wb;
    H1[i][0] = cc * wa * wa; H1[i][1] = cc * wa * wb; H1[i][2] = cc * wb * wb;
  }
  float h2[2], D2[2][2], H2[2][3];
  #pragma unroll
  for (int j = 0; j < 2; ++j) {
    float wa = w2[2 * j], wb = w2[2 * j + 1];
    float z   = wa * h1[0] + wb * h1[1] + b2[j];
    float dz0 = wa * D1[0][0] + wb * D1[1][0];
    float dz1 = wa * D1[0][1] + wb * D1[1][1];
    float dd0 = wa * H1[0][0] + wb * H1[1][0];
    float dd1 = wa * H1[0][1] + wb * H1[1][1];
    float dd2 = wa * H1[0][2] + wb * H1[1][2];
    float h = tanhf(z), g = 1.f - h * h, cc = -2.f * h * g;
    h2[j] = h;
    D2[j][0] = g * dz0; D2[j][1] = g * dz1;
    H2[j][0] = g * dd0 + cc * dz0 * dz0;
    H2[j][1] = g * dd1 + cc * dz0 * dz1;
    H2[j][2] = g * dd2 + cc * dz1 * dz1;
  }
  #pragma unroll
  for (int m = 0; m < 2; ++m) {
    float wa = w3[2 * m], wb = w3[2 * m + 1];
    t[m]       = wa * h2[0]    + wb * h2[1] + b3[m];
    J[2 * m + 0] = wa * D2[0][0] + wb * D2[1][0];
    J[2 * m + 1] = wa * D2[0][1] + wb * D2[1][1];
    H[3 * m + 0] = wa * H2[0][0] + wb * H2[1][0];
    H[3 * m + 1] = wa * H2[0][1] + wb * H2[1][1];
    H[3 * m + 2] = wa * H2[0][2] + wb * H2[1][2];
  }
}

// ---------------- K1: branch conv stack + pooling (128 blocks x 128) -------
__global__ void branch_conv_kernel(const float* __restrict__ f,
    const float* __restrict__ cw1, const float* __restrict__ cb1,
    const float* __restrict__ cw2, const float* __restrict__ cb2,
    const float* __restrict__ cw3, const float* __restrict__ cb3,
    float* __restrict__ feat) {
  __shared__ float buf[2][2][101];
  const int b = blockIdx.x, t = threadIdx.x;
  if (t < 101) {
    buf[0][0][t] = f[b * 202 + t];
    buf[0][1][t] = f[b * 202 + 101 + t];
  }
  __syncthreads();
  const float* cw[3] = {cw1, cw2, cw3};
  const float* cb[3] = {cb1, cb2, cb3};
  int p = 0;
  for (int l = 0; l < 3; ++l) {
    if (t < 101) {
      #pragma unroll
      for (int o = 0; o < 2; ++o) {
        float s = cb[l][o];
        #pragma unroll
        for (int i = 0; i < 2; ++i)
          #pragma unroll
          for (int k = 0; k < 5; ++k) {
            int pos = t - 2 + k;
            float xv = (pos >= 0 && pos < 101) ? buf[p][i][pos] : 0.f;
            s += cw[l][o * 10 + i * 5 + k] * xv;
          }
        buf[1 - p][o][t] = tanhf(s);
      }
    }
    p = 1 - p;
    __syncthreads();
  }
  if (t < 100) {                       // avg-pool pairs over first 100 pos
    int c = t / 50, j = t - c * 50;
    feat[b * 100 + t] = 0.5f * (buf[p][c][2 * j] + buf[p][c][2 * j + 1]);
  }
}

// ---------------- K2: branch FC via WMMA f32 16x16x4 (1 block x 256) ------
// feat(128x100) @ bl1_w^T(100x2, zero-padded to 16 cols) -> tanh -> bl2 -> bl3
__global__ void branch_fc_kernel(const float* __restrict__ feat,
    const float* __restrict__ bl1w, const float* __restrict__ bl1b,
    const float* __restrict__ bl2w, const float* __restrict__ bl2b,
    const float* __restrict__ bl3w, const float* __restrict__ bl3b,
    float* __restrict__ bu, float* __restrict__ bv) {
  __shared__ __align__(16) float Blds[16][100];   // Blds[n][k] = B[k][n], pad 0
  __shared__ float h1[128][2];
  const int tid = threadIdx.x;
  for (int i = tid; i < 16 * 100; i += 256) {
    const int nn = i / 100, kk = i - nn * 100;
    Blds[nn][kk] = (nn < 2) ? bl1w[nn * 100 + kk] : 0.f;
  }
  __syncthreads();
  const int lane = tid & 31, wave = tid >> 5;
  const int n = lane & 15;                 // tile col / A row within tile
  const int m = wave * 16 + n;             // global row
  const int koff = (lane < 16) ? 0 : 2;    // K pair held by this half-wave
  v8f c = {};
  for (int kk = 0; kk < 25; ++kk) {        // K = 100 = 25 * 4, no guards
    const int k0 = kk * 4 + koff;
    v2f a  = *(const v2f*)&feat[m * 100 + k0];   // 8B aligned (k0 even)
    v2f bm = *(const v2f*)&Blds[n][k0];          // 8B aligned LDS
    c = __builtin_amdgcn_wmma_f32_16x16x4_f32(false, a, false, bm,
                                              (short)0, c, false, false);
  }
  if (n < 2) {
    const float bias = bl1b[n];
    const int rbase = wave * 16 + ((lane < 16) ? 0 : 8);
    #pragma unroll
    for (int g = 0; g < 8; ++g) h1[rbase + g][n] = tanhf(c[g] + bias);
  }
  __syncthreads();
  if (tid < 128) {
    float h0 = h1[tid][0], hA = h1[tid][1];
    float a0 = tanhf(bl2w[0] * h0 + bl2w[1] * hA + bl2b[0]);
    float a1 = tanhf(bl2w[2] * h0 + bl2w[3] * hA + bl2b[1]);
    bu[tid] = bl3w[0] * a0 + bl3w[1] * a1 + bl3b[0];
    bv[tid] = bl3w[2] * a0 + bl3w[3] * a1 + bl3b[1];
  }
}

// ---------------- K3: interior sufficient statistics (512 x 256) ----------
__global__ void interior_kernel(const float* __restrict__ xy,
    const float* __restrict__ w1, const float* __restrict__ b1,
    const float* __restrict__ w2, const float* __restrict__ b2,
    const float* __restrict__ w3, const float* __restrict__ b3,
    float* __restrict__ acc) {
  __shared__ float smem[8];
  const int idx = blockIdx.x * blockDim.x + threadIdx.x;   // exactly 131072
  float t[2], J[4], H[6];
  trunk_eval(w1, b1, w2, b2, w3, b3, xy[2 * idx], xy[2 * idx + 1], t, J, H);
  const float c1 = 0.5f * (1.f + NU_C), c2 = 0.5f * (1.f - NU_C);
  float Au = H[0] + c2 * H[2];   // Hu_xx + c2*Hu_yy   (coef of b_u in int_u)
  float Bu = c1 * H[4];          // c1*Hv_xy           (coef of b_v in int_u)
  float Av = c1 * H[1];          // c1*Hu_xy           (coef of b_u in int_v)
  float Bv = H[5] + c2 * H[3];   // Hv_yy + c2*Hv_xx   (coef of b_v in int_v)
  float vals[6] = {Au * Au, Au * Bu, Bu * Bu, Av * Av, Av * Bv, Bv * Bv};
  #pragma unroll
  for (int s = 0; s < 6; ++s) {
    float r = blockReduceSum(vals[s], smem);
    if (threadIdx.x == 0) atomicAdd(&acc[s], r);
  }
}

// ---------------- K4: dirichlet statistics (16 x 256) ---------------------
__global__ void dirichlet_kernel(const float* __restrict__ xy,
    const float* __restrict__ w1, const float* __restrict__ b1,
    const float* __restrict__ w2, const float* __restrict__ b2,
    const float* __restrict__ w3, const float* __restrict__ b3,
    float* __restrict__ acc) {
  __shared__ float smem[8];
  const int idx = blockIdx.x * blockDim.x + threadIdx.x;   // exactly 4096
  float t[2], J[4], H[6];
  trunk_eval(w1, b1, w2, b2, w3, b3, xy[2 * idx], xy[2 * idx + 1], t, J, H);
  float r = blockReduceSum(t[0] * t[0], smem);
  if (threadIdx.x == 0) atomicAdd(&acc[6], r);
  r = blockReduceSum(t[1] * t[1], smem);
  if (threadIdx.x == 0) atomicAdd(&acc[7], r);
}

// ---------------- K5: forcing term, WMMA cross products (1 x 256) ---------
// BT[n][k] (16 x 204, zero padded):
//   rows k=0..100:  n=0 -> cu[k], n=1 -> cv[k]
//   rows k=101..201: n=2 -> du[k-101], n=3 -> dv[k-101]
// R(128x4) = f_batch(128x202) @ B(202x4); WMMA covers k<200, the two
// residual K-columns (200,201) are folded into the final cross-term.
__global__ void forcing_kernel(const float* __restrict__ f,
    const float* __restrict__ xyf,
    const float* __restrict__ w1, const float* __restrict__ b1,
    const float* __restrict__ w2, const float* __restrict__ b2,
    const float* __restrict__ w3, const float* __restrict__ b3,
    const float* __restrict__ bu, const float* __restrict__ bv,
    float* __restrict__ acc) {
  __shared__ __align__(16) float BT[16][204];
  __shared__ float R[128][4];
  __shared__ float smem[8];
  const int tid = threadIdx.x;
  for (int i = tid; i < 16 * 204; i += 256) (&BT[0][0])[i] = 0.f;
  __syncthreads();
  if (tid < 101) {
    float t[2], J[4], H[6];
    trunk_eval(w1, b1, w2, b2, w3, b3, xyf[2 * tid], xyf[2 * tid + 1], t, J, H);
    BT[0][tid]       = PREF_C * J[0];                         // cu
    BT[1][tid]       = PREF_C * NU_C * J[3];                  // cv
    BT[2][101 + tid] = PREF_C * 0.5f * (1.f - NU_C) * J[1];   // du
    BT[3][101 + tid] = PREF_C * 0.5f * (1.f - NU_C) * J[2];   // dv
  }
  float gs = 0.f;                                  // sum of g^2 over f_batch
  for (int i = tid; i < 128 * 202; i += 256) { float g = f[i]; gs += g * g; }
  __syncthreads();
  float cu = 0.f, cv = 0.f, du = 0.f, dv = 0.f;
  if (tid < 101) {
    cu = BT[0][tid]; cv = BT[1][tid];
    du = BT[2][101 + tid]; dv = BT[3][101 + tid];
  }
  float r;
  r = blockReduceSum(cu * cu, smem); if (tid == 0) acc[8]  = r;
  r = blockReduceSum(cu * cv, smem); if (tid == 0) acc[9]  = r;
  r = blockReduceSum(cv * cv, smem); if (tid == 0) acc[10] = r;
  r = blockReduceSum(du * du, smem); if (tid == 0) acc[11] = r;
  r = blockReduceSum(du * dv, smem); if (tid == 0) acc[12] = r;
  r = blockReduceSum(dv * dv, smem); if (tid == 0) acc[13] = r;
  r = blockReduceSum(gs, smem);      if (tid == 0) acc[15] = r;
  __syncthreads();
  const int lane = tid & 31, wave = tid >> 5;
  const int n = lane & 15;
  const int m = wave * 16 + n;
  const int koff = (lane < 16) ? 0 : 2;
  v8f c = {};
  for (int kk = 0; kk < 50; ++kk) {                // k in [0,200), no guards
    const int k0 = kk * 4 + koff;
    v2f a   = *(const v2f*)&f[m * 202 + k0];       // 8B aligned (k0 even)
    v2f bop = *(const v2f*)&BT[n][k0];             // 8B aligned LDS
    c = __builtin_amdgcn_wmma_f32_16x16x4_f32(false, a, false, bop,
                                              (short)0, c, false, false);
  }
  if (n < 4) {
    const int rbase = wave * 16 + ((lane < 16) ? 0 : 8);
    #pragma unroll
    for (int g = 0; g < 8; ++g) R[rbase + g][n] = c[g];
  }
  __syncthreads();
  float term = 0.f;
  if (tid < 128) {
    const float f200 = f[tid * 202 + 200], f201 = f[tid * 202 + 201];
    const float R2 = R[tid][2] + f200 * BT[2][200] + f201 * BT[2][201];
    const float R3 = R[tid][3] + f200 * BT[3][200] + f201 * BT[3][201];
    term = bu[tid] * (R[tid][0] + R2) + bv[tid] * (R[tid][1] + R3);
  }
  r = blockReduceSum(term, smem);
  if (tid == 0) acc[14] = r;
}

// ---------------- K6: combine scalars ----------------
__global__ void finalize_kernel(const float* __restrict__ bu,
                                const float* __restrict__ bv,
                                const float* __restrict__ acc,
                                float* __restrict__ out) {
  if (threadIdx.x == 0 && blockIdx.x == 0) {
    float Su = 0.f, Sv = 0.f, Suv = 0.f;
    for (int i = 0; i < 128; ++i) {
      Su += bu[i] * bu[i]; Sv += bv[i] * bv[i]; Suv += bu[i] * bv[i];
    }
    const float pref2 = PREF_C * PREF_C;
    float li = pref2 * (Su * acc[0] + 2.f * Suv * acc[1] + Sv * acc[2] +
                        Su * acc[3] + 2.f * Suv * acc[4] + Sv * acc[5]) /
               (128.f * 131072.f);
    float ll = (Su * acc[6] + Sv * acc[7]) / (128.f * 4096.f);
    float sf = Su * (acc[8] + acc[11]) + Sv * (acc[10] + acc[13]) +
               2.f * Suv * (acc[9] + acc[12]) + 2.f * acc[14] + acc[15];
    out[0] = li + ll + sf / (128.f * 101.f);
  }
}

// ---------------- host launcher ----------------
extern "C" void kernel_launch(void* const* d_in, const int* in_sizes, int n_in,
                              void* d_out, int out_size, void* d_ws, size_t ws_size,
                              hipStream_t stream) {
  const float* f    = (const float*)d_in[0];   // (128,202)
  const float* xyi  = (const float*)d_in[1];   // (131072,2)
  const float* xyd  = (const float*)d_in[2];   // (4096,2)
  const float* xyf  = (const float*)d_in[3];   // (101,2)

  const float *cw[3], *cb[3], *blw[3], *blb[3], *tlw[3], *tlb[3];
  if (n_in >= 22 && in_sizes[4] == 20) {
    // params flattened in dict insertion order: conv1_w, conv1_b, ...
    cw[0]=(const float*)d_in[4];  cb[0]=(const float*)d_in[5];
    cw[1]=(const float*)d_in[6];  cb[1]=(const float*)d_in[7];
    cw[2]=(const float*)d_in[8];  cb[2]=(const float*)d_in[9];
    blw[0]=(const float*)d_in[10]; blb[0]=(const float*)d_in[11];
    blw[1]=(const float*)d_in[12]; blb[1]=(const float*)d_in[13];
    blw[2]=(const float*)d_in[14]; blb[2]=(const float*)d_in[15];
    tlw[0]=(const float*)d_in[16]; tlb[0]=(const float*)d_in[17];
    tlw[1]=(const float*)d_in[18]; tlb[1]=(const float*)d_in[19];
    tlw[2]=(const float*)d_in[20]; tlb[2]=(const float*)d_in[21];
  } else {
    // jax pytree sorted-key order: bl1_b, bl1_w, ..., conv*_b/w, tl*_b/w
    blb[0]=(const float*)d_in[4];  blw[0]=(const float*)d_in[5];
    blb[1]=(const float*)d_in[6];  blw[1]=(const float*)d_in[7];
    blb[2]=(const float*)d_in[8];  blw[2]=(const float*)d_in[9];
    cb[0]=(const float*)d_in[10];  cw[0]=(const float*)d_in[11];
    cb[1]=(const float*)d_in[12];  cw[1]=(const float*)d_in[13];
    cb[2]=(const float*)d_in[14];  cw[2]=(const float*)d_in[15];
    tlb[0]=(const float*)d_in[16]; tlw[0]=(const float*)d_in[17];
    tlb[1]=(const float*)d_in[18]; tlw[1]=(const float*)d_in[19];
    tlb[2]=(const float*)d_in[20]; tlw[2]=(const float*)d_in[21];
  }

  float* ws   = (float*)d_ws;
  float* feat = ws + WS_FEAT;
  float* bu   = ws + WS_BU;
  float* bv   = ws + WS_BV;
  float* acc  = ws + WS_ACC;

  hipMemsetAsync(acc, 0, 16 * sizeof(float), stream);

  branch_conv_kernel<<<128, 128, 0, stream>>>(f, cw[0], cb[0], cw[1], cb[1],
                                              cw[2], cb[2], feat);
  branch_fc_kernel<<<1, 256, 0, stream>>>(feat, blw[0], blb[0], blw[1], blb[1],
                                          blw[2], blb[2], bu, bv);
  interior_kernel<<<512, 256, 0, stream>>>(xyi, tlw[0], tlb[0], tlw[1], tlb[1],
                                           tlw[2], tlb[2], acc);
  dirichlet_kernel<<<16, 256, 0, stream>>>(xyd, tlw[0], tlb[0], tlw[1], tlb[1],
                                           tlw[2], tlb[2], acc);
  forcing_kernel<<<1, 256, 0, stream>>>(f, xyf, tlw[0], tlb[0], tlw[1], tlb[1],
                                        tlw[2], tlb[2], bu, bv, acc);
  finalize_kernel<<<1, 64, 0, stream>>>(bu, bv, acc, (float*)d_out);
}